// AdaFaceV3_35304631174032
// MI455X (gfx1250) — compile-verified
//
#include <hip/hip_runtime.h>
#include <math.h>

typedef __attribute__((ext_vector_type(2))) float v2f;
typedef __attribute__((ext_vector_type(4))) float v4f;
typedef __attribute__((ext_vector_type(8))) float v8f;

#define AS1 __attribute__((address_space(1)))
#define AS3 __attribute__((address_space(3)))

#if __has_builtin(__builtin_amdgcn_global_load_async_to_lds_b32)
#define HAS_ASYNC 1
#else
#define HAS_ASYNC 0
#endif

#if HAS_ASYNC
# if __has_builtin(__builtin_amdgcn_s_wait_asynccnt)
#  define WAIT_ASYNC() __builtin_amdgcn_s_wait_asynccnt(0)
# else
#  define WAIT_ASYNC() asm volatile("s_wait_asynccnt 0" ::: "memory")
# endif
#else
# define WAIT_ASYNC() ((void)0)
#endif

namespace {
constexpr int   B_    = 1024;
constexpr int   D_    = 512;
constexpr int   C_    = 51332;
constexpr int   KCH_  = 64;          // k-rows staged per LDS stage
constexpr int   LDST_ = 68;          // padded LDS stride (floats): bank = 4*col+k mod 64
constexpr int   NST_  = D_ / KCH_;   // 8 stages
constexpr float EPS_   = 1e-3f;
constexpr float MARG_  = 0.5f;
constexpr float RADH_  = 0.333f;
constexpr float SCALE_ = 64.0f;
constexpr float HEADB_ = 0.5f;
constexpr float BMEAN_ = 0.0f;
constexpr float BSTD_  = 100.0f;
constexpr float PI_    = 3.14159265358979323846f;
}

// ---------------------------------------------------------------------------
// 1) Inverse column L2 norms of kernel [D_, C_] (row-major). Coalesced, and
//    warms the 105 MB kernel matrix into the 192 MB L2 for the GEMM.
// ---------------------------------------------------------------------------
__global__ __launch_bounds__(256)
void colnorm_kernel(const float* __restrict__ kern, float* __restrict__ invn) {
  int c = blockIdx.x * blockDim.x + threadIdx.x;
  if (c >= C_) return;
  float s = 0.0f;
  const float* p = kern + c;
  #pragma unroll 8
  for (int d = 0; d < D_; ++d) {
    float v = p[(size_t)d * C_];
    s = fmaf(v, v, s);
  }
  invn[c] = 1.0f / fmaxf(sqrtf(s), 1e-5f);
}

// Stage one 64-k x 64-col kernel tile into LDS, transposed to lds[col][k]
// (stride LDST_). Wave w covers k-rows w*8..w*8+7; per j, lanes cover 32
// contiguous columns -> 128B-coalesced global reads; per-lane LDS destination
// performs the transpose. Async path tracked on ASYNCcnt.
__device__ __forceinline__
void stage_load(const float* __restrict__ kern, float* __restrict__ buf,
                int k0, int c0, int wave, int lane) {
  #pragma unroll
  for (int j = 0; j < 16; ++j) {
    const int r  = wave * 8 + (j >> 1);          // k-row within stage
    const int c  = ((j & 1) << 5) + lane;        // column within tile
    int cg = c0 + c; if (cg > C_ - 1) cg = C_ - 1;  // clamp ragged tail
    const float* g = kern + (size_t)(k0 + r) * C_ + cg;
    float* l = buf + c * LDST_ + r;
#if HAS_ASYNC
    __builtin_amdgcn_global_load_async_to_lds_b32((AS1 int*)(void*)g,
                                                  (AS3 int*)(void*)l, 0, 0);
#else
    *l = *g;
#endif
  }
}

// ---------------------------------------------------------------------------
// 2) FP32 tensor GEMM (M=1024, N=51332, K=512), V_WMMA_F32_16X16X4_F32.
//    Wave tile 32(M) x 64(N): 2 M-tiles x 4 N-tiles, so every B-fragment
//    ds_load_b128 feeds 4 WMMAs (LDS bandwidth per WMMA halved vs 1 M-tile).
//    Block = 8 waves stacked in M (256 x 64 tile); all waves share the staged
//    B tile. K-slot permutation: double-step of 8 k-rows -> first WMMA uses
//    slots {8d+0,1 | 8d+4,5}, second uses {8d+2,3 | 8d+6,7}, making each
//    lane's A and B data for two WMMAs one contiguous b128.
// ---------------------------------------------------------------------------
__global__ __launch_bounds__(256)
void cos_gemm_kernel(const float* __restrict__ emb, const float* __restrict__ kern,
                     const float* __restrict__ invn, float* __restrict__ out) {
  __shared__ float ldsT[2][64 * LDST_];

  const int lane = threadIdx.x & 31;
  const int wave = threadIdx.x >> 5;
  const int lm   = lane & 15;
  const int khi  = lane >> 4;                    // 0: slots K0/K1, 1: slots K2/K3

  const int mbase = blockIdx.y * 256 + wave * 32;   // two 16-row tiles per wave
  const int c0    = blockIdx.x * 64;

  const float* __restrict__ arow0 = emb + (size_t)(mbase + lm) * D_;
  const float* __restrict__ arow1 = arow0 + (size_t)16 * D_;

  v8f acc[2][4] = {};

  stage_load(kern, ldsT[0], 0, c0, wave, lane);  // prologue: stage 0

  #pragma unroll 1
  for (int s = 0; s < NST_; ++s) {
    WAIT_ASYNC();           // this wave's stage-s copies landed
    __syncthreads();        // all waves' copies landed; prev buffer free
    if (s + 1 < NST_)
      stage_load(kern, ldsT[(s + 1) & 1], (s + 1) * KCH_, c0, wave, lane);

    const float* __restrict__ bp  = ldsT[s & 1];
    const float* __restrict__ ap0 = arow0 + s * KCH_ + 4 * khi;
    const float* __restrict__ ap1 = arow1 + s * KCH_ + 4 * khi;

    #pragma unroll
    for (int dd = 0; dd < KCH_ / 8; ++dd) {
      v4f a0 = *(const v4f*)(ap0 + dd * 8);                      // global b128
      v4f a1 = *(const v4f*)(ap1 + dd * 8);                      // global b128
      #pragma unroll
      for (int t = 0; t < 4; ++t) {
        v4f b4 = *(const v4f*)(bp + (t * 16 + lm) * LDST_ + dd * 8 + 4 * khi); // ds b128
        v2f bl = b4.xy, bh = b4.zw;
        v2f a0l = a0.xy, a0h = a0.zw, a1l = a1.xy, a1h = a1.zw;
        acc[0][t] = __builtin_amdgcn_wmma_f32_16x16x4_f32(
            false, a0l, false, bl, (short)0, acc[0][t], false, false);
        acc[0][t] = __builtin_amdgcn_wmma_f32_16x16x4_f32(
            false, a0h, false, bh, (short)0, acc[0][t], false, false);
        acc[1][t] = __builtin_amdgcn_wmma_f32_16x16x4_f32(
            false, a1l, false, bl, (short)0, acc[1][t], false, false);
        acc[1][t] = __builtin_amdgcn_wmma_f32_16x16x4_f32(
            false, a1h, false, bh, (short)0, acc[1][t], false, false);
      }
    }
    __syncthreads();        // done reading buf[s&1] before it is re-staged
  }

  // Epilogue: cosine = acc * invnorm, clip, scale. cos(acos(x)) == x for all
  // non-label entries, so no transcendentals here. NT stores keep the kernel
  // matrix resident in L2 across the 4 M-blocks.
  #pragma unroll
  for (int t = 0; t < 4; ++t) {
    int c = c0 + 16 * t + lm;
    if (c >= C_) continue;
    float iv = invn[c];
    #pragma unroll
    for (int u = 0; u < 2; ++u) {
      #pragma unroll
      for (int r = 0; r < 8; ++r) {
        int row = mbase + u * 16 + 8 * khi + r;
        float cosv = acc[u][t][r] * iv;
        cosv = fminf(fmaxf(cosv, -1.0f + EPS_), 1.0f - EPS_);
        __builtin_nontemporal_store(cosv * SCALE_, &out[(size_t)row * C_ + c]);
      }
    }
  }
}

// ---------------------------------------------------------------------------
// 3) AdaFace margin fixup at the 1024 label entries (out/S is the clipped
//    cosine, exact since S = 64 is a power of two).
// ---------------------------------------------------------------------------
__global__ __launch_bounds__(256)
void label_fixup_kernel(const float* __restrict__ norms, const int* __restrict__ label,
                        float* __restrict__ out) {
  int b = blockIdx.x * blockDim.x + threadIdx.x;
  if (b >= B_) return;
  int c = label[b];
  float sn = fminf(fmaxf(norms[b], 1e-3f), 100.0f);
  float ms = fminf(fmaxf((sn - BMEAN_) / (BSTD_ + EPS_) * RADH_, -1.0f), 1.0f);
  size_t idx = (size_t)b * C_ + (size_t)c;
  float x = out[idx] * (1.0f / SCALE_);
  float theta = acosf(x) + MARG_ * ms;
  theta = fminf(fmaxf(theta, EPS_), PI_ - EPS_);
  float v = cosf(theta) - (HEADB_ - MARG_ * ms);
  out[idx] = v * SCALE_;
}

extern "C" void kernel_launch(void* const* d_in, const int* in_sizes, int n_in,
                              void* d_out, int out_size, void* d_ws, size_t ws_size,
                              hipStream_t stream) {
  (void)in_sizes; (void)n_in; (void)out_size; (void)ws_size;
  const float* emb   = (const float*)d_in[0];   // [B, D]
  const float* norms = (const float*)d_in[1];   // [B, 1]
  const float* kern  = (const float*)d_in[2];   // [D, C]
  const int*   label = (const int*)d_in[3];     // [B]
  float* out  = (float*)d_out;                  // [B, C]
  float* invn = (float*)d_ws;                   // [C] scratch

  colnorm_kernel<<<(C_ + 255) / 256, 256, 0, stream>>>(kern, invn);

  dim3 grid((C_ + 63) / 64, B_ / 256);          // 803 x 4 blocks of 256 threads
  cos_gemm_kernel<<<grid, dim3(256), 0, stream>>>(emb, kern, invn, out);

  label_fixup_kernel<<<(B_ + 255) / 256, 256, 0, stream>>>(norms, label, out);
}